// StateSpaceModel_23098334117988
// MI455X (gfx1250) — compile-verified
//
#include <hip/hip_runtime.h>

// Kalman filter, B=2048, T=1024, M=4, S=16.
// Cov/gain recursion is batch-independent -> kernel 1 (one wave, all-WMMA)
// precomputes A_t = F^T(I - H^T K_t), B_t = K_t in WMMA B-operand layout;
// kernel 2 runs the batched mean recurrence n <- A_t^T n + B_t^T y^T with
// 5 WMMAs/step per 16-batch tile (memory-bound: ~160MB -> ~7us at 23.3TB/s).
// Both kernels are latency-bound, so K=16 matmuls are tree-accumulated as
// two independent 2-WMMA chains + v_pk_add instead of 4 chained WMMAs.

#define TT 1024
#define BB 2048
#define SS 16
#define MM 4
#define EPSF 1e-4f

typedef float v2f __attribute__((ext_vector_type(2)));
typedef float v8f __attribute__((ext_vector_type(8)));

// V_WMMA_F32_16X16X4_F32: D(16x16) = A(16x4) * B(4x16) + C
__device__ __forceinline__ v8f wmma4(v2f a, v2f b, v8f c) {
  return __builtin_amdgcn_wmma_f32_16x16x4_f32(false, a, false, b, (short)0, c,
                                               false, false);
}

__device__ __forceinline__ v8f vzero() {
  v8f z = {0.f, 0.f, 0.f, 0.f, 0.f, 0.f, 0.f, 0.f};
  return z;
}

// ds_swizzle SWAPX16 (group-of-32, xor=0x10, and=0x1f): swap lane halves.
__device__ __forceinline__ float swap16(float x) {
  int r = __builtin_amdgcn_ds_swizzle(__builtin_bit_cast(int, x), 0x401F);
  return __builtin_bit_cast(float, r);
}

// C/D layout: VGPR g, lane l holds X[g + 8*(l>>4)][l&15].
// B-operand layout chunk c (rows 4c..4c+3), vgpr j: lane l holds
// X[4c + j + 2*(l>>4)][l&15].  Build from C-layout with 8 swizzles + selects.
__device__ __forceinline__ void blayout(const v8f c, bool lo, v2f bch[4]) {
  float s0 = swap16(c[0]), s1 = swap16(c[1]), s2 = swap16(c[2]),
        s3 = swap16(c[3]), s4 = swap16(c[4]), s5 = swap16(c[5]),
        s6 = swap16(c[6]), s7 = swap16(c[7]);
  bch[0] = (v2f){lo ? c[0] : s2, lo ? c[1] : s3};   // rows 0,2 / 1,3
  bch[1] = (v2f){lo ? c[4] : s6, lo ? c[5] : s7};   // rows 4,6 / 5,7
  bch[2] = (v2f){lo ? s0 : c[2], lo ? s1 : c[3]};   // rows 8,10 / 9,11
  bch[3] = (v2f){lo ? s4 : c[6], lo ? s5 : c[7]};   // rows 12,14 / 13,15
}

// 16x16 @ 16x16 as two independent 2-WMMA partial chains + vector add:
// critical path 2 WMMAs instead of 4.
__device__ __forceinline__ v8f mm16(const v2f a[4], const v2f b[4], v8f c0) {
  v8f p0 = wmma4(a[0], b[0], c0);
  v8f p1 = wmma4(a[2], b[2], vzero());
  p0 = wmma4(a[1], b[1], p0);
  p1 = wmma4(a[3], b[3], p1);
  return p0 + p1;
}

// ---------------------------------------------------------------------------
// Kernel 1: one wave runs the whole gain recursion.
//   P in C-layout regs; per step:
//     T1   = F @ P                         (B-op of P via swizzles)
//     covp = T1 @ F^T + Q                  (A-op of T1 via LDS transpose;
//                                           B-layout(F^T) == A-layout(F))
//     HP   = Hpad @ covp
//     S4   = HP H^T + R ; Sinv via cofactors on 16 lanes
//     Kt   = Sinvpad @ HP                  (1 WMMA, K=4)
//     covu = covp - Kt^T @ HP, symmetrize  (1 WMMA + LDS transpose)
//     A_t  = F^T - (HF)^T @ Kt             (1 WMMA)
//   Emits A_t (B-layout, 8 floats/lane) and B_t=Kt (B-layout, 2 floats/lane).
// ---------------------------------------------------------------------------
__global__ __launch_bounds__(32) void k_gains(
    const float* __restrict__ F, const float* __restrict__ H,
    const float* __restrict__ Qc, const float* __restrict__ Rc,
    const float* __restrict__ Pc, float* __restrict__ Abuf,
    float* __restrict__ Bbuf) {
  const int lane = threadIdx.x;
  const int h = lane >> 4;     // lane half
  const int col = lane & 15;
  const bool lo = (lane < 16);

  __shared__ __align__(16) float tr[SS * 20];  // padded transpose buffer
  __shared__ float ldsH[MM * SS];
  __shared__ float hp[MM * SS];
  __shared__ float s4m[16], cofm[16], sim[16];

  for (int i = lane; i < MM * SS; i += 32) ldsH[i] = H[i];

  // --- constant operands in register layouts ---
  v2f fA[4], hA[4];  // A-layout(F), A-layout(Hpad) chunks
#pragma unroll
  for (int k = 0; k < 4; ++k) {
    int kk = 4 * k + 2 * h;
    fA[k] = (v2f){F[col * 16 + kk], F[col * 16 + kk + 1]};
    hA[k] = (col < MM) ? (v2f){H[col * 16 + kk], H[col * 16 + kk + 1]}
                       : (v2f){0.f, 0.f};
  }
  v8f ftC, qC, p;  // C-layouts of F^T, Q = QcQc^T+eps*I, P = PcPc^T+eps*I
#pragma unroll
  for (int g = 0; g < 8; ++g) {
    int row = g + 8 * h;
    ftC[g] = F[col * 16 + row];
    float q = (row == col) ? EPSF : 0.f;
    float pp = q;
    for (int k2 = 0; k2 < 16; ++k2) {
      q += Qc[row * 16 + k2] * Qc[col * 16 + k2];
      pp += Pc[row * 16 + k2] * Pc[col * 16 + k2];
    }
    qC[g] = q;
    p[g] = pp;
  }
  float rv = 0.f;  // R[a][b] for lane<16, a=lane>>2, b=lane&3
  if (lane < 16) {
    int a = lane >> 2, b = lane & 3;
    rv = (a == b) ? EPSF : 0.f;
    for (int k = 0; k < MM; ++k) rv += Rc[a * 4 + k] * Rc[b * 4 + k];
  }
  v2f mhf;  // B-layout(-H@F), single chunk (rows j+2h)
  {
    float x0 = 0.f, x1 = 0.f;
    for (int k = 0; k < 16; ++k) {
      x0 += H[(2 * h) * 16 + k] * F[k * 16 + col];
      x1 += H[(2 * h + 1) * 16 + k] * F[k * 16 + col];
    }
    mhf = (v2f){-x0, -x1};
  }
  __syncthreads();

  for (int t = 0; t < TT; ++t) {
    // T1 = F @ P
    v2f pB[4];
    blayout(p, lo, pB);
    v8f t1 = mm16(fA, pB, vzero());

    // LDS transpose -> A-layout(T1)
#pragma unroll
    for (int g = 0; g < 8; ++g) tr[(g + 8 * h) * 20 + col] = t1[g];
    __syncthreads();
    v2f t1A[4];
#pragma unroll
    for (int k = 0; k < 4; ++k)
      t1A[k] = *(const v2f*)&tr[col * 20 + 4 * k + 2 * h];
    __syncthreads();

    // covp = T1 @ F^T + Q   (B-layout(F^T) == A-layout(F) == fA)
    v8f cp = mm16(t1A, fA, qC);

    // HPfull = Hpad @ covp   (rows 0..3 = H P, rest zero)
    v2f cpB[4];
    blayout(cp, lo, cpB);
    v8f hpv = mm16(hA, cpB, vzero());

    // S4 = HP H^T + R ; invert 4x4 via cofactors (16 lanes)
    if (lo) {
      hp[0 * 16 + col] = hpv[0];
      hp[1 * 16 + col] = hpv[1];
      hp[2 * 16 + col] = hpv[2];
      hp[3 * 16 + col] = hpv[3];
    }
    __syncthreads();
    if (lane < 16) {
      int a = lane >> 2, b = lane & 3;
      float s = rv;
      for (int j = 0; j < 16; ++j) s += hp[a * 16 + j] * ldsH[b * 16 + j];
      s4m[lane] = s;
    }
    __syncthreads();
    if (lane < 16) {
      int a = lane >> 2, b = lane & 3;
      int r0 = (a == 0) ? 1 : 0, r1 = (a <= 1) ? 2 : 1, r2 = (a <= 2) ? 3 : 2;
      int c0 = (b == 0) ? 1 : 0, c1 = (b <= 1) ? 2 : 1, c2 = (b <= 2) ? 3 : 2;
      float m00 = s4m[r0 * 4 + c0], m01 = s4m[r0 * 4 + c1],
            m02 = s4m[r0 * 4 + c2];
      float m10 = s4m[r1 * 4 + c0], m11 = s4m[r1 * 4 + c1],
            m12 = s4m[r1 * 4 + c2];
      float m20 = s4m[r2 * 4 + c0], m21 = s4m[r2 * 4 + c1],
            m22 = s4m[r2 * 4 + c2];
      float d = m00 * (m11 * m22 - m12 * m21) - m01 * (m10 * m22 - m12 * m20) +
                m02 * (m10 * m21 - m11 * m20);
      cofm[lane] = ((a + b) & 1) ? -d : d;
    }
    __syncthreads();
    if (lane < 16) {
      int a = lane >> 2, b = lane & 3;
      float det = s4m[0] * cofm[0] + s4m[1] * cofm[1] + s4m[2] * cofm[2] +
                  s4m[3] * cofm[3];
      sim[lane] = cofm[b * 4 + a] / det;  // Sinv[a][b] = C[b][a]/det
    }
    __syncthreads();

    // Kt = Sinvpad @ HP   (single K=4 chunk)
    v2f siA = (col < MM)
                  ? (v2f){sim[col * 4 + 2 * h], sim[col * 4 + 2 * h + 1]}
                  : (v2f){0.f, 0.f};
    v2f hpB0 = (v2f){lo ? hpv[0] : swap16(hpv[2]),
                     lo ? hpv[1] : swap16(hpv[3])};
    v8f ktv = wmma4(siA, hpB0, vzero());

    // B-layout(Kt) -> also the A-layout of Kt^T (duality)
    v2f ktB = (v2f){lo ? ktv[0] : swap16(ktv[2]),
                    lo ? ktv[1] : swap16(ktv[3])};
    *(v2f*)&Bbuf[t * (4 * SS) + lane * 2] = ktB;  // B_t for kernel 2

    // covu = covp - Kt^T @ HP, then symmetrize
    v2f nktB = (v2f){-ktB[0], -ktB[1]};
    v8f cu = wmma4(nktB, hpB0, cp);
#pragma unroll
    for (int g = 0; g < 8; ++g) tr[(g + 8 * h) * 20 + col] = cu[g];
    __syncthreads();
    {
      const float4 lo4 = *(const float4*)&tr[col * 20 + 8 * h];
      const float4 hi4 = *(const float4*)&tr[col * 20 + 8 * h + 4];
      p[0] = 0.5f * (cu[0] + lo4.x);
      p[1] = 0.5f * (cu[1] + lo4.y);
      p[2] = 0.5f * (cu[2] + lo4.z);
      p[3] = 0.5f * (cu[3] + lo4.w);
      p[4] = 0.5f * (cu[4] + hi4.x);
      p[5] = 0.5f * (cu[5] + hi4.y);
      p[6] = 0.5f * (cu[6] + hi4.z);
      p[7] = 0.5f * (cu[7] + hi4.w);
    }

    // A_t = F^T - (HF)^T @ Kt  (C-layout result)
    v8f at = wmma4(mhf, ktB, ftC);

    // store A_t pre-converted to B-operand layout: 8 floats/lane, 2x b128
    v2f aB[4];
    blayout(at, lo, aB);
    float* ap = &Abuf[t * 256 + lane * 8];
    *(float4*)ap = make_float4(aB[0][0], aB[0][1], aB[1][0], aB[1][1]);
    *(float4*)(ap + 4) = make_float4(aB[2][0], aB[2][1], aB[3][0], aB[3][1]);
    __syncthreads();
  }
}

// ---------------------------------------------------------------------------
// Kernel 2: n = mean^T per 16-batch tile (C/D layout). Per step:
//   n <- A_t^T @ n + B_t^T @ y_t^T
// A-operands load directly (kernel 1 stored B-layout(A_t) == A-layout(A_t^T)),
// B-operand of n built with 8 ds_swizzle + selects; output: 2x b128/lane.
// y-term WMMA seeds one partial accumulator; K=16 matmul split into two
// independent 2-WMMA chains to shorten the serial dependency per step.
// ---------------------------------------------------------------------------
__global__ __launch_bounds__(32) void k_means(
    const float* __restrict__ y, const float* __restrict__ Abuf,
    const float* __restrict__ Bbuf, const float* __restrict__ mean0,
    float* __restrict__ out) {
  const int lane = threadIdx.x;
  const int h = lane >> 4;
  const bool lo = (lane < 16);
  const int b = blockIdx.x * 16 + (lane & 15);  // batch column for this lane

  v8f n;
#pragma unroll
  for (int g = 0; g < 8; ++g) n[g] = mean0[g + 8 * h];

  const float* yp = y + (size_t)b * (TT * MM) + 2 * h;  // y[b][t][2h..2h+1]
  float* op = out + (size_t)b * (TT * SS) + 8 * h;      // out[b][t][8h..8h+7]

  for (int t = 0; t < TT; ++t) {
    const float4 a0 = *(const float4*)&Abuf[t * 256 + lane * 8];
    const float4 a1 = *(const float4*)&Abuf[t * 256 + lane * 8 + 4];
    const v2f kb = *(const v2f*)&Bbuf[t * (4 * SS) + lane * 2];
    const v2f yv = *(const v2f*)&yp[(size_t)t * MM];
    if (t + 8 < TT)
      __builtin_prefetch(&Abuf[(t + 8) * 256 + lane * 8], 0, 0);

    v8f c0 = wmma4(kb, yv, vzero());  // B_t^T @ y_t^T (off critical path)

    float s0 = swap16(n[0]), s1 = swap16(n[1]), s2 = swap16(n[2]),
          s3 = swap16(n[3]), s4 = swap16(n[4]), s5 = swap16(n[5]),
          s6 = swap16(n[6]), s7 = swap16(n[7]);
    v2f B0 = (v2f){lo ? n[0] : s2, lo ? n[1] : s3};
    v2f B1 = (v2f){lo ? n[4] : s6, lo ? n[5] : s7};
    v2f B2 = (v2f){lo ? s0 : n[2], lo ? s1 : n[3]};
    v2f B3 = (v2f){lo ? s4 : n[6], lo ? s5 : n[7]};

    v8f c1 = wmma4((v2f){a0.x, a0.y}, B0, c0);
    v8f c2 = wmma4((v2f){a1.x, a1.y}, B2, vzero());
    c1 = wmma4((v2f){a0.z, a0.w}, B1, c1);
    c2 = wmma4((v2f){a1.z, a1.w}, B3, c2);
    n = c1 + c2;

    *(float4*)(op + (size_t)t * SS) = make_float4(n[0], n[1], n[2], n[3]);
    *(float4*)(op + (size_t)t * SS + 4) = make_float4(n[4], n[5], n[6], n[7]);
  }
}

extern "C" void kernel_launch(void* const* d_in, const int* in_sizes, int n_in,
                              void* d_out, int out_size, void* d_ws,
                              size_t ws_size, hipStream_t stream) {
  const float* y = (const float*)d_in[0];    // [B,T,M]
  const float* F = (const float*)d_in[1];    // [S,S]
  const float* H = (const float*)d_in[2];    // [M,S]
  const float* Qc = (const float*)d_in[3];   // [S,S]
  const float* Rc = (const float*)d_in[4];   // [M,M]
  const float* m0 = (const float*)d_in[5];   // [S]
  const float* Pc = (const float*)d_in[6];   // [S,S]

  float* Abuf = (float*)d_ws;                 // T * 256 floats (B-layout A_t)
  float* Bbuf = Abuf + (size_t)TT * 256;      // T * 64 floats  (B-layout K_t)

  k_gains<<<1, 32, 0, stream>>>(F, H, Qc, Rc, Pc, Abuf, Bbuf);
  k_means<<<BB / 16, 32, 0, stream>>>(y, Abuf, Bbuf, m0, (float*)d_out);
}